// Sequence_59554016526400
// MI455X (gfx1250) — compile-verified
//
#include <hip/hip_runtime.h>
#include <hip/hip_bf16.h>

// Problem constants (from reference)
#define T_LEN 262144
#define HN 40      // hidden size
#define GN 160     // 4*H gates
#define VN 20      // vocab
#define EN 30      // embed dim

typedef float v2f __attribute__((ext_vector_type(2)));
typedef float v8f __attribute__((ext_vector_type(8)));

__device__ __forceinline__ float sigmoid_f(float x) {
    return 1.f / (1.f + __expf(-x));
}
__device__ __forceinline__ float tanh_f(float x) {
    // numerically safe: saturates to +/-1 for large |x|
    float e = __expf(2.f * x);
    return 1.f - 2.f / (e + 1.f);
}
__device__ __forceinline__ float readlane_f(float v, int lane) {
    return __int_as_float(__builtin_amdgcn_readlane(__float_as_int(v), lane));
}

// ---------------------------------------------------------------------------
// Phase 1: per-vocab pre-activation tables.
// table_d[v][g] = b_ih_d[g] + b_hh_d[g] + sum_e W_ih_d[g][e] * embed[v][e]
// Tokens only take 20 values -> the whole T x 160 x 30 GEMM collapses to this.
// ---------------------------------------------------------------------------
__global__ void table_kernel(const float* __restrict__ embed,
                             const float* __restrict__ W_ih1,
                             const float* __restrict__ b_ih1,
                             const float* __restrict__ b_hh1,
                             const float* __restrict__ W_ih2,
                             const float* __restrict__ b_ih2,
                             const float* __restrict__ b_hh2,
                             float* __restrict__ tables) {
    const int d = blockIdx.x;
    const float* W  = d ? W_ih2 : W_ih1;
    const float* bi = d ? b_ih2 : b_ih1;
    const float* bh = d ? b_hh2 : b_hh1;
    float* tab = tables + d * (VN * GN);
    const int g = threadIdx.x;                 // 0..159
    const float bsum = bi[g] + bh[g];
    for (int v = 0; v < VN; ++v) {
        float acc = bsum;
        #pragma unroll
        for (int e = 0; e < EN; ++e)
            acc = __builtin_fmaf(W[g * EN + e], embed[v * EN + e], acc);
        tab[v * GN + g] = acc;
    }
}

// ---------------------------------------------------------------------------
// Phase 2: the serial LSTM scans. One wave32 per direction (2 blocks).
// Lane L owns gates {L, 32+L, 64+L, 96+L, 128+L}; W_hh rows live in VGPRs
// (200 floats/lane), h broadcast via v_readlane (constant lane after unroll).
// Gate->unit regrouping goes through a 160-float LDS scratch once per step.
// ---------------------------------------------------------------------------
__global__ __launch_bounds__(32)
void lstm_scan_kernel(const int* __restrict__ tokens,
                      const float* __restrict__ W_hh1,
                      const float* __restrict__ W_hh2,
                      const float* __restrict__ tables,
                      float* __restrict__ Hbuf) {
    const int d = blockIdx.x;            // 0 = forward, 1 = backward
    const int L = threadIdx.x;           // 0..31
    const float* W    = d ? W_hh2 : W_hh1;
    const float* tabg = tables + d * (VN * GN);
    float* Hd = Hbuf + (size_t)d * (size_t)T_LEN * HN;

    __shared__ float tab[VN * GN];       // 12.8 KB pre-activation table
    __shared__ float glds[GN];           // per-step gate scratch

    for (int i = L; i < VN * GN; i += 32) tab[i] = tabg[i];

    // W_hh rows for this lane's 5 gates -> registers: w[j][k] = W[(j*32+L)][k]
    float w[5][HN];
    #pragma unroll
    for (int j = 0; j < 5; ++j)
        #pragma unroll
        for (int k = 0; k < HN; ++k)
            w[j][k] = W[(j * 32 + L) * HN + k];

    __syncthreads();

    // h state: ha = h[L] (all lanes), hb = h[32+L] (valid on lanes 0..7)
    float ha = 0.f, hb = 0.f, ca = 0.f, cb = 0.f;

    for (int tb = 0; tb < T_LEN; tb += 32) {
        // preload 32 tokens for this chunk into a register, broadcast later
        const int tidx  = d ? (T_LEN - 1 - (tb + L)) : (tb + L);
        const int tok32 = tokens[tidx];

        #pragma unroll 1
        for (int s = 0; s < 32; ++s) {
            const int tok = __builtin_amdgcn_readlane(tok32, s);
            const float* xr = &tab[tok * GN];
            float a0 = xr[L];
            float a1 = xr[32 + L];
            float a2 = xr[64 + L];
            float a3 = xr[96 + L];
            float a4 = xr[128 + L];
            // gates += h @ W_hh.T  (h broadcast via readlane, weights in VGPRs)
            #pragma unroll
            for (int k = 0; k < HN; ++k) {
                const float hk = readlane_f(k < 32 ? ha : hb, k & 31);
                a0 = __builtin_fmaf(hk, w[0][k], a0);
                a1 = __builtin_fmaf(hk, w[1][k], a1);
                a2 = __builtin_fmaf(hk, w[2][k], a2);
                a3 = __builtin_fmaf(hk, w[3][k], a3);
                a4 = __builtin_fmaf(hk, w[4][k], a4);
            }
            // regroup gate-major -> unit-major through LDS
            glds[L]       = a0;
            glds[32 + L]  = a1;
            glds[64 + L]  = a2;
            glds[96 + L]  = a3;
            glds[128 + L] = a4;
            __syncthreads();   // single-wave WG: waitcnt + NOP barrier

            // unit n1 = L : gate layout [i | f | g | o] each length 40
            {
                const float ig = sigmoid_f(glds[L]);
                const float fg = sigmoid_f(glds[40 + L]);
                const float gg = tanh_f(glds[80 + L]);
                const float og = sigmoid_f(glds[120 + L]);
                ca = fg * ca + ig * gg;
                ha = og * tanh_f(ca);
            }
            if (L < 8) {       // unit n2 = 32 + L
                const float ig = sigmoid_f(glds[32 + L]);
                const float fg = sigmoid_f(glds[72 + L]);
                const float gg = tanh_f(glds[112 + L]);
                const float og = sigmoid_f(glds[152 + L]);
                cb = fg * cb + ig * gg;
                hb = og * tanh_f(cb);
            }
            const int t = tb + s;
            Hd[(size_t)t * HN + L] = ha;
            if (L < 8) Hd[(size_t)t * HN + 32 + L] = hb;
            __syncthreads();   // protect glds WAR before next step's stores
        }
    }
}

// ---------------------------------------------------------------------------
// Phase 3: out = H1 @ W_l1.T + H2 @ W_l2.T + (b_l1 + b_l2), via
// V_WMMA_F32_16X16X4_F32. One wave per 16-row tile; two 16-col tiles
// (cols >= V are zero-padded / store-masked). K=40 -> 10 chunks of 4,
// x2 matrices = 20 WMMAs per output tile.
// A (16x4 f32): lane M=L&15; vgpr0/1 hold K = {0,1} (lanes 0-15), {2,3}
// (lanes 16-31). B (4x16) mirrored; C/D: vgpr r -> row r (lanes 0-15) or
// row 8+r (lanes 16-31), col = L&15.
// ---------------------------------------------------------------------------
__global__ __launch_bounds__(32)
void proj_kernel(const float* __restrict__ Hbuf,
                 const float* __restrict__ W_l1,
                 const float* __restrict__ b_l1,
                 const float* __restrict__ W_l2,
                 const float* __restrict__ b_l2,
                 float* __restrict__ out) {
    const int L     = threadIdx.x;
    const int t0    = blockIdx.x * 16;
    const int r16   = L & 15;
    const int khalf = L >> 4;                       // 0 or 1
    const float* H1 = Hbuf;
    const float* H2 = Hbuf + (size_t)T_LEN * HN;

    #pragma unroll
    for (int c = 0; c < 2; ++c) {
        const int  n  = c * 16 + r16;               // output vocab column
        const bool nv = (n < VN);
        v8f acc = {};

        #pragma unroll
        for (int kc = 0; kc < 10; ++kc) {
            const int kb = kc * 4 + khalf * 2;
            v2f a, b;
            const float* ap = &H1[(size_t)(t0 + r16) * HN + kb];
            a.x = ap[0]; a.y = ap[1];
            if (nv) { b.x = W_l1[n * HN + kb]; b.y = W_l1[n * HN + kb + 1]; }
            else    { b.x = 0.f;               b.y = 0.f; }
            acc = __builtin_amdgcn_wmma_f32_16x16x4_f32(
                false, a, false, b, (short)0, acc, false, false);
        }
        #pragma unroll
        for (int kc = 0; kc < 10; ++kc) {
            const int kb = kc * 4 + khalf * 2;
            v2f a, b;
            const float* ap = &H2[(size_t)(t0 + r16) * HN + kb];
            a.x = ap[0]; a.y = ap[1];
            if (nv) { b.x = W_l2[n * HN + kb]; b.y = W_l2[n * HN + kb + 1]; }
            else    { b.x = 0.f;               b.y = 0.f; }
            acc = __builtin_amdgcn_wmma_f32_16x16x4_f32(
                false, a, false, b, (short)0, acc, false, false);
        }

        const float bias = nv ? (b_l1[n] + b_l2[n]) : 0.f;
        #pragma unroll
        for (int r = 0; r < 8; ++r) {
            const int m = (L < 16) ? r : (8 + r);
            if (nv) out[(size_t)(t0 + m) * VN + n] = acc[r] + bias;
        }
    }
}

// ---------------------------------------------------------------------------
extern "C" void kernel_launch(void* const* d_in, const int* in_sizes, int n_in,
                              void* d_out, int out_size, void* d_ws, size_t ws_size,
                              hipStream_t stream) {
    (void)in_sizes; (void)n_in; (void)out_size; (void)ws_size;
    const int*   tokens = (const int*)  d_in[0];
    const float* embed  = (const float*)d_in[1];
    const float* W_ih1  = (const float*)d_in[2];
    const float* W_hh1  = (const float*)d_in[3];
    const float* b_ih1  = (const float*)d_in[4];
    const float* b_hh1  = (const float*)d_in[5];
    const float* W_ih2  = (const float*)d_in[6];
    const float* W_hh2  = (const float*)d_in[7];
    const float* b_ih2  = (const float*)d_in[8];
    const float* b_hh2  = (const float*)d_in[9];
    const float* W_l1   = (const float*)d_in[10];
    const float* b_l1   = (const float*)d_in[11];
    const float* W_l2   = (const float*)d_in[12];
    const float* b_l2   = (const float*)d_in[13];

    float* ws     = (float*)d_ws;
    float* tables = ws;          // 2 * 20 * 160 floats
    float* Hbuf   = ws + 8192;   // 2 * T * 40 floats (H1 then H2)

    table_kernel<<<2, GN, 0, stream>>>(embed, W_ih1, b_ih1, b_hh1,
                                       W_ih2, b_ih2, b_hh2, tables);
    lstm_scan_kernel<<<2, 32, 0, stream>>>(tokens, W_hh1, W_hh2, tables, Hbuf);
    proj_kernel<<<T_LEN / 16, 32, 0, stream>>>(Hbuf, W_l1, b_l1, W_l2, b_l2,
                                               (float*)d_out);
}